// DAELoss_68152541053132
// MI455X (gfx1250) — compile-verified
//
#include <hip/hip_runtime.h>
#include <hip/hip_bf16.h>

#define BB 512
#define SS 128
#define VV 2048
#define NEG_HUGE (-3.402823466e+38f)

#if defined(__has_builtin)
#if __has_builtin(__builtin_amdgcn_wmma_f32_16x16x4_f32)
#define HAVE_WMMA_F32X4 1
#endif
#endif

typedef __attribute__((ext_vector_type(2))) float v2f;
typedef __attribute__((ext_vector_type(4))) float v4f;
typedef __attribute__((ext_vector_type(8))) float v8f;

// ---------------------------------------------------------------------------
// Kernel 1: one wave32 per (b,s) row of V=2048 logits. Single streaming pass:
// online softmax (max, sum-exp), sum(x), x[target], argmax (first occurrence).
// ce[b,s] = lse - 0.9*x_t - 0.1*mean(x)  (zeroed at PAD), pred[b,s] = argmax.
// ---------------------------------------------------------------------------
__global__ __launch_bounds__(256) void k_rowstats(const float* __restrict__ x,
                                                  const int* __restrict__ tgt,
                                                  float* __restrict__ ce,
                                                  int* __restrict__ pred)
{
    const int wave = threadIdx.x >> 5;
    const int lane = threadIdx.x & 31;
    const int row  = blockIdx.x * 8 + wave;

    const v4f* xr = (const v4f*)(x + (size_t)row * VV);
    const int t = tgt[row];

    float m = NEG_HUGE, z = 0.0f, sx = 0.0f, xt = 0.0f;
    float amv = NEG_HUGE; int ami = 0;

#pragma unroll 4
    for (int i = 0; i < 16; ++i) {
        const v4f* p = xr + (size_t)i * 32 + lane;
        v4f v = __builtin_nontemporal_load(p);      // global_load_b128 (NT)
        const int base = (i * 32 + lane) * 4;
#pragma unroll
        for (int k = 0; k < 4; ++k) {
            const float cv = v[k];
            const int idx = base + k;
            sx += cv;
            if (cv > amv) { amv = cv; ami = idx; }      // ascending idx -> first occurrence
            if (cv > m)   { z = z * __expf(m - cv) + 1.0f; m = cv; }
            else          { z += __expf(cv - m); }
            if (idx == t) xt = cv;                       // exactly one (lane,i,k) owns t
        }
    }

    // wave32 combine: log-sum-exp merge + sum + argmax-with-tiebreak
#pragma unroll
    for (int off = 16; off > 0; off >>= 1) {
        float m2 = __shfl_xor(m, off, 32);
        float z2 = __shfl_xor(z, off, 32);
        float M  = fmaxf(m, m2);
        z = z * __expf(m - M) + z2 * __expf(m2 - M);
        m = M;
        sx += __shfl_xor(sx, off, 32);
        xt += __shfl_xor(xt, off, 32);
        float av2 = __shfl_xor(amv, off, 32);
        int   ai2 = __shfl_xor(ami, off, 32);
        if (av2 > amv || (av2 == amv && ai2 < ami)) { amv = av2; ami = ai2; }
    }

    if (lane == 0) {
        const float lse = m + __logf(z);
        ce[row]  = (t != 0) ? (lse - 0.9f * xt - 0.1f * (sx * (1.0f / VV))) : 0.0f;
        pred[row] = ami;
    }
}

// ---------------------------------------------------------------------------
// Kernel 2: one block per batch row. Computes lens/plen, analytic position
// weights, weighted-CE partials, length penalty, bigram/trigram counts.
// partials[b*8 + {0:numer,1:denom,2:lenpen,3:bigram,4:trigram,5:anyvalid}]
// ---------------------------------------------------------------------------
__global__ __launch_bounds__(128) void k_perbatch(const float* __restrict__ ce,
                                                  const int* __restrict__ pred,
                                                  const int* __restrict__ tgt,
                                                  float* __restrict__ partials)
{
    __shared__ int   sp[SS];
    __shared__ int   st[SS];
    __shared__ int   icnt[2];
    __shared__ float facc[6];

    const int b = blockIdx.x;
    const int s = threadIdx.x;
    const int row = b * SS + s;

    if (s < 2) icnt[s] = 0;
    if (s < 6) facc[s] = 0.0f;

    const int p = pred[row];
    const int t = tgt[row];
    sp[s] = p;
    st[s] = t;
    const float cev = ce[row];
    if (b + 1 < BB) __builtin_prefetch(ce + row + SS, 0, 0);   // global_prefetch_b8
    __syncthreads();

    if (t != 0) atomicAdd(&icnt[0], 1);
    if (p != 0) atomicAdd(&icnt[1], 1);
    __syncthreads();

    const int L = icnt[0];   // lens[b]
    const int P = icnt[1];   // plen[b]

    // analytic position-weight row W[L-1][s], then where(s < L, ., 1.0)
    float bw = 1.0f;
    if (s < L) {
        float w = 1.0f + 0.5f * ((float)s / (float)L);
        if (s == L - 1)            w = 4.5f;   // END_W*1.5
        if (L >= 2 && s == L - 2)  w = 3.0f;   // END_W*1.0
        if (L >= 3 && s == L - 3)  w = 2.4f;   // END_W*0.8
        if (L >= 4 && s >= L / 3 && s < (2 * L) / 3) w *= 1.3f;
        if (L <= 4)                w *= 1.2f;
        bw = w;
    }
    atomicAdd(&facc[0], cev * bw);   // numerator
    atomicAdd(&facc[1], bw);         // denominator

    if (s == 0) {
        const float diff = fabsf((float)P - (float)L);
        const float factor = 1.0f + (P < L ? 0.5f : 0.0f) + (P <= 3 ? 0.3f : 0.0f);
        atomicAdd(&facc[2], diff * factor);
    }
    if (t != 0 && s < SS - 2) facc[5] = 1.0f;   // any_valid flag (benign race)

    if (s < SS - 1) {   // bigram
        const bool pb = (sp[s] == sp[s + 1]);
        const bool tb = (st[s] == st[s + 1]);
        const bool mb = pb && tb && (sp[s] == st[s]);
        const float w2 = (s >= SS - 3) ? 2.25f : 1.0f;   // 1.5^2 on last two
        const float cnt = (pb ? 1.0f : 0.0f) + (tb ? 1.0f : 0.0f) - (mb ? 2.0f : 0.0f);
        if (cnt != 0.0f) atomicAdd(&facc[3], cnt * w2);
    }
    if (s < SS - 2) {   // trigram
        const bool pb0 = (sp[s] == sp[s + 1]), pb1 = (sp[s + 1] == sp[s + 2]);
        const bool tb0 = (st[s] == st[s + 1]), tb1 = (st[s + 1] == st[s + 2]);
        const bool pt = pb0 && pb1, tt = tb0 && tb1;
        const bool mt = pt && tt && (sp[s] == st[s]);
        const float w2 = (s >= SS - 4) ? 4.0f : 1.0f;    // 2.0^2 on last two
        const float cnt = (pt ? 1.0f : 0.0f) + (tt ? 1.0f : 0.0f) - (mt ? 2.0f : 0.0f);
        if (cnt != 0.0f) atomicAdd(&facc[4], cnt * w2);
    }
    __syncthreads();
    if (s < 6) partials[(size_t)b * 8 + s] = facc[s];
}

// ---------------------------------------------------------------------------
// Kernel 3: 6 waves; wave w sums its 512 partials via v_wmma_f32_16x16x4_f32
// with a ones B-matrix (D[m,n] = sum_k A[m,k], accumulated over 8 tiles).
// Then compose the final scalar.
// ---------------------------------------------------------------------------
__global__ __launch_bounds__(192) void k_final(const float* __restrict__ partials,
                                               float* __restrict__ out)
{
    __shared__ float res[6];
    const int wave = threadIdx.x >> 5;   // 0..5 -> which partial quantity
    const int lane = threadIdx.x & 31;

    float total;
#if defined(HAVE_WMMA_F32X4)
    {
        // A 16x4 f32 layout: lanes 0-15 hold M=lane (K=0,1); lanes 16-31 hold
        // M=lane-16 (K=2,3). Ones B => D[m,n] = sum_k A[m,k] (+C accumulate).
        const int mrow  = lane & 15;
        const int kbase = (lane >> 4) * 2;
        v8f c = {};
        v2f bones; bones.x = 1.0f; bones.y = 1.0f;
#pragma unroll
        for (int tile = 0; tile < 8; ++tile) {
            const int i0 = tile * 64 + mrow * 4 + kbase;
            v2f a;
            a.x = partials[(size_t)i0 * 8 + wave];
            a.y = partials[(size_t)(i0 + 1) * 8 + wave];
            c = __builtin_amdgcn_wmma_f32_16x16x4_f32(false, a, false, bones,
                                                      (short)0, c, false, false);
        }
        // D layout: VGPR r = row M=r (lanes 0-15) / M=r+8 (lanes 16-31).
        float sum8 = c[0] + c[1] + c[2] + c[3] + c[4] + c[5] + c[6] + c[7];
        sum8 += __shfl_xor(sum8, 16, 32);   // fold M=0..7 with M=8..15
        total = sum8;
    }
#else
    {
        float acc = 0.0f;
        for (int i = lane; i < BB; i += 32) acc += partials[(size_t)i * 8 + wave];
#pragma unroll
        for (int off = 16; off > 0; off >>= 1) acc += __shfl_xor(acc, off, 32);
        total = acc;
    }
#endif
    if (lane == 0) res[wave] = total;
    __syncthreads();

    if (threadIdx.x == 0) {
        const float numer = res[0], denom = res[1], lensum = res[2];
        const float bsum = res[3], tsum = res[4], anyv = res[5];
        const float weighted = numer / denom;
        const float lenpen   = 0.3f * (lensum * (1.0f / (float)BB));
        const float bigram   = bsum / ((float)BB * (float)(SS - 1) * (float)VV);
        const float trigram  = tsum / ((float)BB * (float)(SS - 2) * (float)VV);
        const float ngram    = bigram + ((anyv > 0.0f) ? 1.5f * trigram : 0.0f);
        out[0] = 0.7f * weighted + 0.2f * lenpen + 0.1f * 0.2f * ngram;  // CHAR_W=0.2
    }
}

// ---------------------------------------------------------------------------
extern "C" void kernel_launch(void* const* d_in, const int* in_sizes, int n_in,
                              void* d_out, int out_size, void* d_ws, size_t ws_size,
                              hipStream_t stream) {
    (void)in_sizes; (void)n_in; (void)out_size; (void)ws_size;
    const float* x   = (const float*)d_in[0];
    const int*   tgt = (const int*)d_in[1];

    float* ce       = (float*)d_ws;                                   // B*S floats
    int*   pred     = (int*)((char*)d_ws + (size_t)BB * SS * 4);      // B*S ints
    float* partials = (float*)((char*)d_ws + (size_t)BB * SS * 8);    // B*8 floats

    k_rowstats<<<(BB * SS) / 8, 256, 0, stream>>>(x, tgt, ce, pred);
    k_perbatch<<<BB, SS, 0, stream>>>(ce, pred, tgt, partials);
    k_final<<<1, 192, 0, stream>>>(partials, (float*)d_out);
}